// InfoNCECritic_4123168604380
// MI455X (gfx1250) — compile-verified
//
#include <hip/hip_runtime.h>
#include <math.h>

// ---------------------------------------------------------------------------
// InfoNCE critic for MI455X (gfx1250, wave32).
//  Phase 1: xp = x@W1[:64], yp = y@W1[64:]  via V_WMMA_F32_16X16X4_F32 (fp32).
//  Phase 2: pairwise scores s[i,j] = (sum_k relu(yp[i,k]+xp[j,k]+b1[k])*w2[k]
//           + b2)/TEMP. Grid = (64 i-tiles x 16 j-slices) for occupancy.
//           xp chunks staged with double-buffered GLOBAL_LOAD_ASYNC_TO_LDS_B128
//           (ASYNCcnt), streaming logsumexp partials per (row, slice).
//           pos_i == s[i,i] captured on the fly.
//  Phase 3: merge 16 LSE partials per row, then
//           loss = -(mean(pos) - mean(lse) + log n).
// ---------------------------------------------------------------------------

typedef __attribute__((ext_vector_type(2))) float v2f;
typedef __attribute__((ext_vector_type(8))) float v8f;

#define NROWS 1024
#define DIN   64
#define HID   256
#define STR   260          // LDS row stride (floats): 16B-aligned, conflict-free
#define NSLICE 16          // j-slices per row
#define CPS    4           // 16-j chunks per slice (64 j per block)
#define INV_TEMP 10.0f

// ---------------- Phase 1: fp32 WMMA GEMM --------------------------------
// One wave (32 threads) per 16x16 output tile. K=64 in steps of 4.
__global__ __launch_bounds__(32) void gemm16x16x4_f32(
    const float* __restrict__ x, const float* __restrict__ y,
    const float* __restrict__ W1,
    float* __restrict__ xp, float* __restrict__ yp) {
  const int lane   = threadIdx.x;
  const int lane16 = lane & 15;
  const int half   = lane >> 4;          // 0 or 1
  const int n0 = blockIdx.x * 16;
  const int m0 = blockIdx.y * 16;
  const int sel = blockIdx.z;            // 0: x path, 1: y path

  const float* src = sel ? y : x;
  const float* W   = W1 + sel * DIN * HID;   // W1x rows 0..63, W1y rows 64..127
  float*       dst = sel ? yp : xp;

  const float* arow = src + (m0 + lane16) * DIN + 2 * half;
  const float* bcol = W + (2 * half) * HID + (n0 + lane16);

  v8f c = {};
  #pragma unroll
  for (int k = 0; k < DIN; k += 4) {
    v2f a = *(const v2f*)(arow + k);         // global_load_b64, 8B aligned
    v2f b;
    b.x = bcol[k * HID];                     // row k + 2*half
    b.y = bcol[k * HID + HID];               // row k + 2*half + 1
    c = __builtin_amdgcn_wmma_f32_16x16x4_f32(
        /*neg_a=*/false, a, /*neg_b=*/false, b,
        /*c_mod=*/(short)0, c, /*reuse_a=*/false, /*reuse_b=*/false);
  }

  #pragma unroll
  for (int r = 0; r < 8; ++r) {
    const int row = m0 + r + 8 * half;
    dst[row * HID + n0 + lane16] = c[r];
  }
}

// ---------------- Phase 2: pairwise scores + streaming LSE partials -------
// Grid: (64, 16) blocks, 256 threads (8 waves). Thread t owns
// (il = t>>4, jl = t&15) and sweeps 4 chunks of 16 j within its slice.
__global__ __launch_bounds__(256) void score_lse_part(
    const float* __restrict__ xp, const float* __restrict__ yp,
    const float* __restrict__ b1, const float* __restrict__ w2,
    const float* __restrict__ b2p,
    float* __restrict__ pos, float* __restrict__ pm, float* __restrict__ ps) {
  __shared__ __align__(16) float v[16 * STR];       // yp tile + b1
  __shared__ __align__(16) float xq[2][16 * STR];   // double-buffered xp chunks
  __shared__ __align__(16) float w2s[HID];
  __shared__ float redm[256];
  __shared__ float reds[256];

  const int t  = threadIdx.x;
  const int i0 = blockIdx.x * 16;
  const int sb = blockIdx.y;                 // j-slice index
  const int il = t >> 4;
  const int jl = t & 15;

  // Stage yp i-tile (+b1) and w2 into LDS (regular path, once per block).
  for (int idx = t; idx < 16 * HID; idx += 256) {
    const int row = idx >> 8, col = idx & 255;
    v[row * STR + col] = yp[(i0 + row) * HID + col] + b1[col];
  }
  w2s[t] = w2[t];
  const float b2v = *b2p;

  // Issue async direct-to-LDS load of chunk 0 (16 rows x 256 floats).
  {
    const int cb = sb * CPS;
    #pragma unroll
    for (int q = 0; q < 4; ++q) {
      const int idx = t + q * 256;
      const int row = idx >> 6, c4 = (idx & 63) * 4;
      const unsigned loff = (unsigned)(uintptr_t)(&xq[0][row * STR + c4]);
      const float* g = xp + (size_t)(cb * 16 + row) * HID + c4;
      asm volatile("global_load_async_to_lds_b128 %0, %1, off"
                   :: "v"(loff), "v"(g) : "memory");
    }
  }

  float m = -INFINITY, ssum = 0.0f;
  for (int c = 0; c < CPS; ++c) {
    asm volatile("s_wait_asynccnt 0x0" ::: "memory");  // own async issues done
    __syncthreads();                                   // whole chunk visible

    if (c + 1 < CPS) {                    // prefetch next chunk into other buf
      const int cb = sb * CPS + c + 1;
      #pragma unroll
      for (int q = 0; q < 4; ++q) {
        const int idx = t + q * 256;
        const int row = idx >> 6, c4 = (idx & 63) * 4;
        const unsigned loff =
            (unsigned)(uintptr_t)(&xq[(c + 1) & 1][row * STR + c4]);
        const float* g = xp + (size_t)(cb * 16 + row) * HID + c4;
        asm volatile("global_load_async_to_lds_b128 %0, %1, off"
                     :: "v"(loff), "v"(g) : "memory");
      }
    }

    const float* __restrict__ vv = &v[il * STR];
    const float* __restrict__ xx = &xq[c & 1][jl * STR];
    float acc = 0.0f;
    #pragma unroll 4
    for (int k = 0; k < HID; k += 4) {     // 3x ds_load_b128, 12 VALU ops
      const float4 a = *(const float4*)(vv + k);
      const float4 b = *(const float4*)(xx + k);
      const float4 w = *(const float4*)(w2s + k);
      acc = fmaf(fmaxf(a.x + b.x, 0.0f), w.x, acc);
      acc = fmaf(fmaxf(a.y + b.y, 0.0f), w.y, acc);
      acc = fmaf(fmaxf(a.z + b.z, 0.0f), w.z, acc);
      acc = fmaf(fmaxf(a.w + b.w, 0.0f), w.w, acc);
    }
    const float score = (acc + b2v) * INV_TEMP;

    const int j = (sb * CPS + c) * 16 + jl;
    if (j == i0 + il) pos[i0 + il] = score;        // pos_i == scores[i,i]

    const float nm = fmaxf(m, score);              // streaming logsumexp
    ssum = ssum * __expf(m - nm) + __expf(score - nm);
    m = nm;
  }

  // Merge the 16 (m,ssum) partials that share row il (threads il*16..il*16+15).
  redm[t] = m; reds[t] = ssum;
  __syncthreads();
  if (jl == 0) {
    float M = redm[t], S = reds[t];
    #pragma unroll
    for (int q = 1; q < 16; ++q) {
      const float m2 = redm[t + q], s2 = reds[t + q];
      const float nm = fmaxf(M, m2);
      S = S * __expf(M - nm) + s2 * __expf(m2 - nm);
      M = nm;
    }
    pm[(i0 + il) * NSLICE + sb] = M;
    ps[(i0 + il) * NSLICE + sb] = S;
  }
}

// ---------------- Phase 3: merge slice partials + final reduction ---------
__global__ __launch_bounds__(256) void finalize(
    const float* __restrict__ pos, const float* __restrict__ pm,
    const float* __restrict__ ps, float* __restrict__ out) {
  __shared__ float sp[256];
  __shared__ float sl[256];
  const int t = threadIdx.x;
  float ap = 0.0f, al = 0.0f;
  for (int i = t; i < NROWS; i += 256) {
    float M = pm[i * NSLICE], S = ps[i * NSLICE];
    #pragma unroll
    for (int q = 1; q < NSLICE; ++q) {
      const float m2 = pm[i * NSLICE + q], s2 = ps[i * NSLICE + q];
      const float nm = fmaxf(M, m2);
      S = S * __expf(M - nm) + s2 * __expf(m2 - nm);
      M = nm;
    }
    al += M + __logf(S);
    ap += pos[i];
  }
  sp[t] = ap; sl[t] = al;
  __syncthreads();
  for (int s = 128; s > 0; s >>= 1) {
    if (t < s) { sp[t] += sp[t + s]; sl[t] += sl[t + s]; }
    __syncthreads();
  }
  if (t == 0) {
    const float n = (float)NROWS;
    const float lower_bound = sp[0] / n - (sl[0] / n - __logf(n));
    out[0] = -lower_bound;
  }
}

// ---------------------------------------------------------------------------
extern "C" void kernel_launch(void* const* d_in, const int* in_sizes, int n_in,
                              void* d_out, int out_size, void* d_ws, size_t ws_size,
                              hipStream_t stream) {
  const float* x  = (const float*)d_in[0];
  const float* y  = (const float*)d_in[1];
  const float* W1 = (const float*)d_in[2];
  const float* b1 = (const float*)d_in[3];
  const float* w2 = (const float*)d_in[4];
  const float* b2 = (const float*)d_in[5];
  float* out = (float*)d_out;

  float* xp  = (float*)d_ws;              // 1024*256
  float* yp  = xp + NROWS * HID;          // 1024*256
  float* pos = yp + NROWS * HID;          // 1024
  float* pm  = pos + NROWS;               // 1024*16
  float* ps  = pm + NROWS * NSLICE;       // 1024*16

  gemm16x16x4_f32<<<dim3(HID / 16, NROWS / 16, 2), 32, 0, stream>>>(x, y, W1, xp, yp);
  score_lse_part<<<dim3(NROWS / 16, NSLICE), 256, 0, stream>>>(
      xp, yp, b1, w2, b2, pos, pm, ps);
  finalize<<<1, 256, 0, stream>>>(pos, pm, ps, out);
}